// SubSeSoap_50989851738336
// MI455X (gfx1250) — compile-verified
//
#include <hip/hip_runtime.h>
#include <math.h>

typedef __attribute__((ext_vector_type(2))) float v2f;
typedef __attribute__((ext_vector_type(8))) float v8f;

#define RC_F 6.0f
#define ATOM_STRIDE 40              // floats per atom in LDS (2*40 % 64 == 16 -> bank decorrelated)
#define WAVE_LDS (32 * ATOM_STRIDE) // 1280 floats per wave

__global__ void soap_zero_c(float* __restrict__ c) {
    c[threadIdx.x] = 0.0f;
}

__global__ __launch_bounds__(256) void soap_accum(const float* __restrict__ coo,
                                                  const int* __restrict__ numbers,
                                                  float* __restrict__ cbuf, int n) {
    __shared__ float lds[8 * WAVE_LDS]; // 8 waves * 1280 floats = 40 KB
    const int lane = threadIdx.x & 31;
    const int wave = threadIdx.x >> 5;
    const int lid  = lane & 15;
    const int half = lane >> 4;
    float* W = lds + wave * WAVE_LDS;

    v8f acc = {}; // D = 16x16 f32 accumulator; rows 0-7 = c[(s,n),(l,k)]

    for (long base = (long)blockIdx.x * 256; base < (long)n; base += (long)gridDim.x * 256) {
        const long atom = base + (long)wave * 32 + lane;
        float x = 0.0f, y = 0.0f, z = 0.0f;
        int spec = 0;
        if (atom < (long)n) {
            x = coo[3 * atom + 0];
            y = coo[3 * atom + 1];
            z = coo[3 * atom + 2];
            spec = numbers[atom];
        }
        // radial part: f[n] = rad * exp(-d^2/2) * r2^n
        const float r2 = x * x + y * y + z * z;
        const float d  = sqrtf(r2);
        const float t1 = 1.0f - d * (1.0f / RC_F);
        float rr = ((d < RC_F) ? t1 * t1 : 0.0f) * __expf(-0.5f * r2);
        if (atom >= (long)n) rr = 0.0f; // dead lanes contribute zero
        const float f0 = rr, f1 = rr * r2, f2 = f1 * r2, f3 = f2 * r2;

        // packed solid harmonics (lmax=3)
        const float Re00 = 0.28209479177387814f; // 0.5/sqrt(pi)
        const float Re11 = -1.2247448713915890f * x * Re00;
        const float Im11 = -1.2247448713915890f * y * Re00;
        const float Re22 = -1.1180339887498949f * (x * Re11 - y * Im11);
        const float Im22 = -1.1180339887498949f * (x * Im11 + y * Re11);
        const float Re33 = -1.0801234497346435f * (x * Re22 - y * Im22);
        const float Im33 = -1.0801234497346435f * (x * Im22 + y * Re22);
        const float Re10 = 1.7320508075688772f * z * Re00;
        const float Re21 = 2.2360679774997896f * z * Re11;
        const float Im21 = 2.2360679774997896f * z * Im11;
        const float Re32 = 2.6457513110645906f * z * Re22;
        const float Im32 = 2.6457513110645906f * z * Im22;
        const float Re20 = 1.9364916731037085f * z * Re10 - 1.1180339887498949f * r2 * Re00;
        const float Re30 = 1.9720265943665387f * z * Re20 - 1.0183501544346312f * r2 * Re10;
        const float Re31 = 2.0916500663351889f * z * Re21 - 0.9354143466934853f * r2 * Re11;
        const float Im31 = 2.0916500663351889f * z * Im21 - 0.9354143466934853f * r2 * Im11;

        // stage: Y[0..15], A[0..7] (species one-hot x f), zeros[8]
        float4* dst = (float4*)(W + lane * ATOM_STRIDE);
        dst[0] = make_float4(Re00, Im11, Im22, Im33);
        dst[1] = make_float4(Re11, Re10, Im21, Im32);
        dst[2] = make_float4(Re22, Re21, Re20, Im31);
        dst[3] = make_float4(Re33, Re32, Re31, Re30);
        const bool s0 = (spec == 0);
        dst[4] = make_float4(s0 ? f0 : 0.0f, s0 ? f1 : 0.0f, s0 ? f2 : 0.0f, s0 ? f3 : 0.0f);
        dst[5] = make_float4(s0 ? 0.0f : f0, s0 ? 0.0f : f1, s0 ? 0.0f : f2, s0 ? 0.0f : f3);
        dst[6] = make_float4(0.0f, 0.0f, 0.0f, 0.0f);
        dst[7] = make_float4(0.0f, 0.0f, 0.0f, 0.0f);
        __syncthreads();

        // 8 WMMAs retire the 32-atom tile: D += A(16x4) * B(4x16)
        // A layout: VGPR0 = K (atom) {0 | 2}, VGPR1 = {1 | 3}; lanes = M rows.
        // B layout: same K split; lanes = N columns.
#pragma unroll
        for (int g = 0; g < 8; ++g) {
            const float* p0 = W + (g * 4 + half * 2) * ATOM_STRIDE;
            const float* p1 = p0 + ATOM_STRIDE;
            v2f A, B;
            A.x = p0[16 + lid];
            A.y = p1[16 + lid];
            B.x = p0[lid];
            B.y = p1[lid];
            acc = __builtin_amdgcn_wmma_f32_16x16x4_f32(
                /*neg_a=*/false, A, /*neg_b=*/false, B,
                /*c_mod=*/(short)0, acc, /*reuse_a=*/false, /*reuse_b=*/false);
        }
        __syncthreads();
    }

    // block-level reduction of the 8 wave accumulators (rows 0-7 live in lanes 0-15)
    if (half == 0) {
#pragma unroll
        for (int j = 0; j < 8; ++j)
            lds[wave * 128 + j * 16 + lid] = acc[j];
    }
    __syncthreads();
    if (threadIdx.x < 128) {
        float s = 0.0f;
#pragma unroll
        for (int w = 0; w < 8; ++w)
            s += lds[w * 128 + threadIdx.x];
        atomicAdd(&cbuf[threadIdx.x], s); // global_atomic_add_f32
    }
}

__device__ __forceinline__ float soap_aconst(int n, int l) {
    const float fact[7] = {1.f, 1.f, 2.f, 6.f, 24.f, 120.f, 720.f};
    return exp2f(-(float)(2 * n + l)) / ((float)(2 * l + 1) * fact[n] * fact[n + l]);
}

__global__ __launch_bounds__(256) void soap_finalize(const float* __restrict__ cbuf,
                                                     float* __restrict__ out) {
    __shared__ float cs[128];
    __shared__ float ps[256];
    __shared__ float red[256];
    const int t = threadIdx.x;
    if (t < 128) cs[t] = cbuf[t];
    __syncthreads();

    // p[a,b,n,m,j] = sum_k Yr[j,k]*c[a,n,j,k]*c[b,m,j,k] + sum_l Yi[l,j]*c[a,n,l,j]*c[b,m,l,j]
    const int j = t & 3, m = (t >> 2) & 3, nn = (t >> 4) & 3, b = (t >> 6) & 1, a = (t >> 7) & 1;
    const float* ca = cs + (a * 4 + nn) * 16;
    const float* cb = cs + (b * 4 + m) * 16;
    float term = 0.0f;
    for (int k = 0; k <= j; ++k)
        term += ((k < j) ? 2.0f : 1.0f) * ca[j * 4 + k] * cb[j * 4 + k];
    for (int l = 0; l < j; ++l)
        term += 2.0f * ca[l * 4 + j] * cb[l * 4 + j];
    const float p = term * sqrtf(soap_aconst(nn, j) * soap_aconst(m, j));

    ps[t] = p;
    red[t] = p * p;
    __syncthreads();
    for (int s = 128; s > 0; s >>= 1) {
        if (t < s) red[t] += red[t + s];
        __syncthreads();
    }
    const float inv = 1.0f / (sqrtf(red[0]) + 1.1920928955078125e-07f);
    out[t] = ps[t] * inv;
}

extern "C" void kernel_launch(void* const* d_in, const int* in_sizes, int n_in,
                              void* d_out, int out_size, void* d_ws, size_t ws_size,
                              hipStream_t stream) {
    const float* coo = (const float*)d_in[0];
    const int* numbers = (const int*)d_in[1];
    float* out = (float*)d_out;
    float* cbuf = (float*)d_ws; // 128-float species accumulator
    const int n = in_sizes[1];

    soap_zero_c<<<1, 128, 0, stream>>>(cbuf);
    soap_accum<<<512, 256, 0, stream>>>(coo, numbers, cbuf, n);
    soap_finalize<<<1, 256, 0, stream>>>(cbuf, out);
}